// HanningTemplateLayer_61761629716908
// MI455X (gfx1250) — compile-verified
//
#include <hip/hip_runtime.h>
#include <hip/hip_bf16.h>

typedef __attribute__((ext_vector_type(2))) float v2f;
typedef __attribute__((ext_vector_type(4))) float v4f;
typedef __attribute__((ext_vector_type(8))) float v8f;

#define L_LEN   65536
#define N_ROWS  64
#define TAPS    80          // combined filter support: offsets -40..39
#define KSPAN   96          // 24 chunks of K=4 cover u in [0,95]
#define NCHUNK  (KSPAN / 4)
#define PAD     16          // zero padding around filter in LDS
#define WAVES_PER_BLOCK 8
#define TILES_TOTAL (N_ROWS * (L_LEN / 256))   // 16384 waves, 256 outputs each

// out[row, j] = sum_{t=0}^{79} eff[t] * x[row, j-40+t]   (zero padded)
// Tiled as D[m,n] = out[row, j0 + m + 16n] = sum_u A[m,u] B[u,n]
//   A[m,u] = eff[u-m]                (16x96 Toeplitz, register-resident)
//   B[u,n] = x[row, j0-40+u+16n]    (96x16 signal slice)
__global__ __launch_bounds__(256) void
HanningTemplateLayer_61761629716908_kernel(const float* __restrict__ x,
                                           const float* __restrict__ tw,
                                           float* __restrict__ out) {
    __shared__ float eff_pad[128];   // [0..15]=0, [16..95]=eff[0..79], [96..127]=0

    // ---- build combined filter (softmax-weighted sum of 4 Hann windows) ----
    {
        const int t = threadIdx.x;
        if (t < 128) {
            float acc = 0.0f;
            const int tt = t - PAD;               // filter tap index
            if ((unsigned)tt < TAPS) {
                float w0 = tw[0], w1 = tw[1], w2 = tw[2], w3 = tw[3];
                float mx = fmaxf(fmaxf(w0, w1), fmaxf(w2, w3));
                float e0 = __expf(w0 - mx), e1 = __expf(w1 - mx);
                float e2 = __expf(w2 - mx), e3 = __expf(w3 - mx);
                float inv = 1.0f / (e0 + e1 + e2 + e3);
                const float sw[4]   = {e0 * inv, e1 * inv, e2 * inv, e3 * inv};
                const int   wid4[4] = {10, 20, 30, 40};
                const float twopi = 6.28318530717958647692f;
                #pragma unroll
                for (int i = 0; i < 4; ++i) {
                    int w = wid4[i];
                    int k = tt - 40 + w;          // index into hanning(2w)
                    if (k >= 0 && k < 2 * w) {
                        float h = 0.5f - 0.5f * cosf(twopi * (float)k / (float)(2 * w - 1));
                        acc += sw[i] * h;
                    }
                }
            }
            eff_pad[t] = acc;
        }
    }
    __syncthreads();

    // ---- per-wave tile assignment ----
    const int lane = threadIdx.x & 31;
    const int wid  = threadIdx.x >> 5;
    const int W    = blockIdx.x * WAVES_PER_BLOCK + wid;   // 0..16383
    const int row  = W >> 8;                               // 0..63
    const int tile = W & 255;                              // 0..255
    const int j0   = tile << 8;                            // 256 outputs per wave

    const int m   = lane & 15;   // A-row (output sub-pos) == B-col n
    const int hi  = lane >> 4;   // lane-group: K element pair select
    const int klo = hi << 1;

    // ---- A fragments: A[m,u] = eff[u-m]; padded LDS -> unconditional ds loads ----
    v2f a_frag[NCHUNK];
    #pragma unroll
    for (int c = 0; c < NCHUNK; ++c) {
        const int base = c * 4 + klo - m + PAD;   // in [1, 111] -- always in bounds
        a_frag[c].x = eff_pad[base];
        a_frag[c].y = eff_pad[base + 1];
    }

    // ---- 24 fp32 WMMAs over K = 96, split into two independent accumulator
    //      chains (even/odd chunks) to halve the WMMA RAW dependency depth ----
    const float* xrow  = x + (size_t)row * L_LEN;
    const int    basep = j0 - 40 + 16 * m + klo;  // signal pos of chunk0 elem .x (even)
    v8f acc0 = {};
    v8f acc1 = {};

    const bool interior = (j0 >= 40) && (j0 + 296 <= L_LEN);
    if (interior) {
        #pragma unroll
        for (int c = 0; c < NCHUNK; c += 2) {
            v2f b0 = *(const v2f*)(xrow + basep + 4 * c);       // 8B-aligned b64
            v2f b1 = *(const v2f*)(xrow + basep + 4 * (c + 1));
            acc0 = __builtin_amdgcn_wmma_f32_16x16x4_f32(
                false, a_frag[c], false, b0, (short)0, acc0, false, false);
            acc1 = __builtin_amdgcn_wmma_f32_16x16x4_f32(
                false, a_frag[c + 1], false, b1, (short)0, acc1, false, false);
        }
    } else {
        #pragma unroll
        for (int c = 0; c < NCHUNK; c += 2) {
            const int p0 = basep + 4 * c;                   // always even
            const int p1 = p0 + 4;
            int pc0 = p0 < 0 ? 0 : p0;                      // clamp keeps 8B alignment
            pc0 = pc0 > (L_LEN - 2) ? (L_LEN - 2) : pc0;
            int pc1 = p1 < 0 ? 0 : p1;
            pc1 = pc1 > (L_LEN - 2) ? (L_LEN - 2) : pc1;
            v2f b0 = *(const v2f*)(xrow + pc0);             // unconditional b64 load
            v2f b1 = *(const v2f*)(xrow + pc1);
            b0.x = ((unsigned)p0       < L_LEN) ? b0.x : 0.0f;  // v_cndmask
            b0.y = ((unsigned)(p0 + 1) < L_LEN) ? b0.y : 0.0f;
            b1.x = ((unsigned)p1       < L_LEN) ? b1.x : 0.0f;
            b1.y = ((unsigned)(p1 + 1) < L_LEN) ? b1.y : 0.0f;
            acc0 = __builtin_amdgcn_wmma_f32_16x16x4_f32(
                false, a_frag[c], false, b0, (short)0, acc0, false, false);
            acc1 = __builtin_amdgcn_wmma_f32_16x16x4_f32(
                false, a_frag[c + 1], false, b1, (short)0, acc1, false, false);
        }
    }

    // ---- combine chains; store D element r -> out[row, j0 + (r + 8*hi) + 16*m]
    //      (8 consecutive floats per lane -> two coalesced b128 stores) ----
    float* orow = out + (size_t)row * L_LEN + j0 + 8 * hi + 16 * m;
    v4f lo = {acc0[0] + acc1[0], acc0[1] + acc1[1], acc0[2] + acc1[2], acc0[3] + acc1[3]};
    v4f hh = {acc0[4] + acc1[4], acc0[5] + acc1[5], acc0[6] + acc1[6], acc0[7] + acc1[7]};
    *(v4f*)(orow)     = lo;   // 16B-aligned global_store_b128
    *(v4f*)(orow + 4) = hh;
}

extern "C" void kernel_launch(void* const* d_in, const int* in_sizes, int n_in,
                              void* d_out, int out_size, void* d_ws, size_t ws_size,
                              hipStream_t stream) {
    (void)in_sizes; (void)n_in; (void)d_ws; (void)ws_size; (void)out_size;
    const float* x  = (const float*)d_in[0];   // [64, 65536] fp32
    const float* tw = (const float*)d_in[1];   // [4] fp32
    float* out = (float*)d_out;                // [64, 65536] fp32

    const int blocks = TILES_TOTAL / WAVES_PER_BLOCK;      // 2048
    HanningTemplateLayer_61761629716908_kernel<<<blocks, 256, 0, stream>>>(x, tw, out);
}